// GCN_En_29755533426825
// MI455X (gfx1250) — compile-verified
//
#include <hip/hip_runtime.h>
#include <hip/hip_bf16.h>

typedef __attribute__((ext_vector_type(2))) float v2f;
typedef __attribute__((ext_vector_type(8))) float v8f;

#define NFEAT 256
#define NHID  64

// ---------------------------------------------------------------------------
// Kernel 1: support = x @ W   via V_WMMA_F32_16X16X4_F32 (exact fp32 math;
// GEMM is bandwidth-bound so the f32 WMMA path is strictly the right choice).
// Each wave owns a 16(M) x 64(N) tile: 4 v8f accumulators, K-loop of 64 steps.
// W is staged transposed into LDS with an XOR swizzle so B-fragment reads are
// 8B-aligned, bank-conflict-free ds_load_b64s. 256 threads = 8 waves = 128 rows.
// ---------------------------------------------------------------------------
__global__ __launch_bounds__(256) void gcn_gemm_wmma_f32(
    const float* __restrict__ x, const float* __restrict__ W,
    float* __restrict__ support, int n_nodes)
{
  __shared__ float Wt[NHID * NFEAT];  // 64 * 256 * 4 = 64 KB, XOR-swizzled

  const int tid = threadIdx.x;
  // Stage W[k][n] -> Wt[n*256 + (k ^ 4*(n&15))]
  for (int i = tid; i < NFEAT * NHID; i += 256) {
    int k = i >> 6;         // 0..255
    int n = i & 63;         // 0..63
    Wt[n * NFEAT + (k ^ ((n & 15) << 2))] = W[i];
  }
  __syncthreads();

  const int wave = tid >> 5;
  const int lane = tid & 31;
  const int half = lane >> 4;   // 0: K=k0,k0+1   1: K=k0+2,k0+3
  const int l15  = lane & 15;

  const long rowBase = (long)blockIdx.x * 128 + (long)wave * 16;
  int arow = (int)rowBase + l15;
  if (arow >= n_nodes) arow = n_nodes - 1;   // clamp loads; EXEC stays all-1s
  const float* xrow = x + (long)arow * NFEAT;

  const float* b0p = &Wt[( 0 + l15) * NFEAT];
  const float* b1p = &Wt[(16 + l15) * NFEAT];
  const float* b2p = &Wt[(32 + l15) * NFEAT];
  const float* b3p = &Wt[(48 + l15) * NFEAT];
  const int swz = l15 << 2;

  v8f acc0 = {}, acc1 = {}, acc2 = {}, acc3 = {};

  for (int k0 = 0; k0 < NFEAT; k0 += 4) {
    const int k  = k0 + (half << 1);   // even -> 8B aligned
    const int kx = k ^ swz;            // XOR swizzle preserves the pair
    v2f a  = *(const v2f*)(xrow + k);
    v2f b0 = *(const v2f*)(b0p + kx);
    v2f b1 = *(const v2f*)(b1p + kx);
    v2f b2 = *(const v2f*)(b2p + kx);
    v2f b3 = *(const v2f*)(b3p + kx);
    acc0 = __builtin_amdgcn_wmma_f32_16x16x4_f32(false, a, false, b0, (short)0, acc0, false, false);
    acc1 = __builtin_amdgcn_wmma_f32_16x16x4_f32(false, a, false, b1, (short)0, acc1, false, false);
    acc2 = __builtin_amdgcn_wmma_f32_16x16x4_f32(false, a, false, b2, (short)0, acc2, false, false);
    acc3 = __builtin_amdgcn_wmma_f32_16x16x4_f32(false, a, false, b3, (short)0, acc3, false, false);
  }

  // C/D layout: VGPR v, lanes 0-15 -> M=v, lanes 16-31 -> M=v+8, N=lane%16
#pragma unroll
  for (int v = 0; v < 8; ++v) {
    long m = rowBase + v + 8 * half;
    if (m < n_nodes) {
      float* dst = support + m * NHID + l15;
      dst[ 0] = acc0[v];
      dst[16] = acc1[v];
      dst[32] = acc2[v];
      dst[48] = acc3[v];
    }
  }
}

// ---------------------------------------------------------------------------
// Kernel 2: edge scatter.  agg[dst] += support[src] * w  with hardware
// global_atomic_add_f32.  16 threads per edge, float4 gathers: support and
// agg (25.6 MB each) live in the 192 MB L2, so this stays on-die.
// ---------------------------------------------------------------------------
__global__ __launch_bounds__(256) void gcn_edge_scatter(
    const float* __restrict__ support, const int* __restrict__ esrc,
    const int* __restrict__ edst, const float* __restrict__ ew,
    float* __restrict__ agg, int n_edges)
{
  long i = (long)blockIdx.x * 256 + threadIdx.x;
  long e = i >> 4;
  if (e >= n_edges) return;
  int f = ((int)i & 15) << 2;
  int s = esrc[e];
  int d = edst[e];
  float w = ew[e];
  const float4 v = *(const float4*)(support + (long)s * NHID + f);
  float* out = agg + (long)d * NHID + f;
  unsafeAtomicAdd(out + 0, v.x * w);
  unsafeAtomicAdd(out + 1, v.y * w);
  unsafeAtomicAdd(out + 2, v.z * w);
  unsafeAtomicAdd(out + 3, v.w * w);
}

// ---------------------------------------------------------------------------
// Kernel 3: h = relu(agg + b); inverted dropout with JAX threefry2x32,
// key(42) -> (k0,k1)=(0,42); element i pairs with i+n/2 (JAX iota split).
// ---------------------------------------------------------------------------
__device__ __forceinline__ unsigned rotl32(unsigned x, int r) {
  return (x << r) | (x >> (32 - r));
}

__global__ __launch_bounds__(256) void gcn_finalize(
    float* __restrict__ out, const float* __restrict__ bias, int half)
{
  int i = blockIdx.x * 256 + threadIdx.x;
  if (i >= half) return;
  int j = i + half;

  const unsigned ks0 = 0u, ks1 = 42u, ks2 = 0u ^ 42u ^ 0x1BD11BDAu;
  unsigned x0 = (unsigned)i + ks0;
  unsigned x1 = (unsigned)j + ks1;
#define TF_R(r) { x0 += x1; x1 = rotl32(x1, r); x1 ^= x0; }
  TF_R(13) TF_R(15) TF_R(26) TF_R(6)
  x0 += ks1; x1 += ks2 + 1u;
  TF_R(17) TF_R(29) TF_R(16) TF_R(24)
  x0 += ks2; x1 += ks0 + 2u;
  TF_R(13) TF_R(15) TF_R(26) TF_R(6)
  x0 += ks0; x1 += ks1 + 3u;
  TF_R(17) TF_R(29) TF_R(16) TF_R(24)
  x0 += ks1; x1 += ks2 + 4u;
#undef TF_R

  float u0 = __uint_as_float((x0 >> 9) | 0x3f800000u) - 1.0f;
  float u1 = __uint_as_float((x1 >> 9) | 0x3f800000u) - 1.0f;

  float h0 = out[i] + bias[i & 63];
  float h1 = out[j] + bias[j & 63];
  h0 = fmaxf(h0, 0.0f);
  h1 = fmaxf(h1, 0.0f);
  out[i] = (u0 < 0.5f) ? h0 * 2.0f : 0.0f;
  out[j] = (u1 < 0.5f) ? h1 * 2.0f : 0.0f;
}

// ---------------------------------------------------------------------------
extern "C" void kernel_launch(void* const* d_in, const int* in_sizes, int n_in,
                              void* d_out, int out_size, void* d_ws, size_t ws_size,
                              hipStream_t stream) {
  const float* x    = (const float*)d_in[0];
  const float* W    = (const float*)d_in[1];
  const float* bias = (const float*)d_in[2];
  const int*   esrc = (const int*)d_in[3];
  const int*   edst = (const int*)d_in[4];
  const float* ew   = (const float*)d_in[5];
  float* out = (float*)d_out;
  float* support = (float*)d_ws;   // [n_nodes * NHID] fp32

  const int n_nodes = in_sizes[0] / NFEAT;   // 100000
  const int n_edges = in_sizes[3];           // 1600000

  // d_out doubles as the scatter accumulator: zero it first.
  hipMemsetAsync(out, 0, (size_t)out_size * sizeof(float), stream);

  // GEMM: 128 rows per workgroup (8 waves x 16 rows).
  int gblocks = (n_nodes + 127) / 128;
  gcn_gemm_wmma_f32<<<gblocks, 256, 0, stream>>>(x, W, support, n_nodes);

  // Scatter: 16 threads per edge.
  long swork = (long)n_edges * 16;
  int sblocks = (int)((swork + 255) / 256);
  gcn_edge_scatter<<<sblocks, 256, 0, stream>>>(support, esrc, edst, ew, out, n_edges);

  // Bias + ReLU + threefry dropout (processes index pairs (i, i+half)).
  int half = out_size / 2;
  int fblocks = (half + 255) / 256;
  gcn_finalize<<<fblocks, 256, 0, stream>>>(out, bias, half);
}